// Net3_9887014715535
// MI455X (gfx1250) — compile-verified
//
#include <hip/hip_runtime.h>
#include <math.h>

#define EPS     1e-8f
#define INFEAT  4096
#define CAP     65536
#define WAVESZ  32
#define WAVES   8
#define TPB     (WAVES * WAVESZ)   // 256 threads = 8 wave32

typedef float v2f __attribute__((ext_vector_type(2)));
typedef float v8f __attribute__((ext_vector_type(8)));

// ---------------------------------------------------------------------------
// Kernel 1: streaming GEMV + row sum-of-squares in one pass over memory.
// Each wave computes a 16-row tile via V_WMMA_F32_16X16X4_F32:
//   A (16x4 f32): lane L (0-15) = row L, lanes 16-31 carry the other K half.
//   B (4x16 f32): x-chunk replicated across all 16 columns -> every column of
//                 D holds the same running dot (dot products are K-permutation
//                 invariant, so any consistent half/VGPR->K map is correct).
// Norms are accumulated on the VALU from the same loaded registers.
// Writes sims' = dot / max(||mem_row||, eps) to simsOut (x-norm folded later).
// ---------------------------------------------------------------------------
__global__ __launch_bounds__(TPB) void gemv_cos_kernel(
    const float* __restrict__ x,
    const float* __restrict__ mem,
    float* __restrict__ simsOut)
{
    __shared__ __align__(16) float sx[INFEAT];

    const int tid = threadIdx.x;
    // Stage x into LDS (coalesced b128 loads, 16 KB).
    for (int i = tid * 4; i < INFEAT; i += TPB * 4) {
        *(float4*)&sx[i] = *(const float4*)&x[i];
    }
    __syncthreads();

    const int lane = tid & (WAVESZ - 1);
    const int wave = tid >> 5;
    const int r    = lane & 15;        // row within tile (== WMMA M for this lane)
    const int h    = lane >> 4;        // which K-half this lane carries
    const int tile = blockIdx.x * WAVES + wave;
    const int row  = tile * 16 + r;

    // Lane reads 4 consecutive floats of its row per 8-k step:
    // float offset = kb + 4*h  ->  float4 index kb/4 + h.
    const float4* pm = (const float4*)(mem + (size_t)row * INFEAT) + h;

    v8f   acc = {0.f, 0.f, 0.f, 0.f, 0.f, 0.f, 0.f, 0.f};
    float nsq = 0.f;

#pragma unroll 4
    for (int k4 = 0; k4 < INFEAT / 4; k4 += 2) {
        const float4 m  = pm[k4];                              // global_load_b128
        const float4 xb = *(const float4*)&sx[k4 * 4 + 4 * h]; // ds_load_b128

        v2f a0; a0.x = m.x;  a0.y = m.y;
        v2f b0; b0.x = xb.x; b0.y = xb.y;
        acc = __builtin_amdgcn_wmma_f32_16x16x4_f32(
                  false, a0, false, b0, (short)0, acc, false, false);

        v2f a1; a1.x = m.z;  a1.y = m.w;
        v2f b1; b1.x = xb.z; b1.y = xb.w;
        acc = __builtin_amdgcn_wmma_f32_16x16x4_f32(
                  false, a1, false, b1, (short)0, acc, false, false);

        nsq = fmaf(m.x, m.x, nsq);
        nsq = fmaf(m.y, m.y, nsq);
        nsq = fmaf(m.z, m.z, nsq);
        nsq = fmaf(m.w, m.w, nsq);
    }

    // Row sum-of-squares: lane L and lane L+16 each hold half the K range.
    const float nsqFull = nsq + __shfl(nsq, lane | 16);

    // Extract dots from the documented C/D layout:
    //   rows 0-7  -> acc[row]   in lanes 0-15  (all N columns equal)
    //   rows 8-15 -> acc[row-8] in lanes 16-31
    const int src = (r < 8) ? 0 : 16;
    float t[8];
#pragma unroll
    for (int i = 0; i < 8; ++i) t[i] = __shfl(acc[i], src);
    float dot = t[0];
#pragma unroll
    for (int i = 1; i < 8; ++i) if ((r & 7) == i) dot = t[i];

    if (lane < 16) {
        const float mn = fmaxf(sqrtf(nsqFull), EPS);
        simsOut[tile * 16 + lane] = dot / mn;
    }
}

// ---------------------------------------------------------------------------
// Kernel 2: single block. Reduce ||x||, argmax over 65536 sims'
// (first-index tie-break like jnp.argmax), emit {idx, sims'[idx]/||x||c}.
// ---------------------------------------------------------------------------
__global__ __launch_bounds__(1024) void argmax_kernel(
    const float* __restrict__ x,
    const float* __restrict__ sims,
    float* __restrict__ res)   // res[0] = idx bits, res[1] = value
{
    __shared__ float sv[1024];
    __shared__ int   si[1024];
    const int tid = threadIdx.x;

    // ||x||^2
    float s = 0.f;
    for (int i = tid; i < INFEAT; i += 1024) {
        const float v = x[i];
        s = fmaf(v, v, s);
    }
    sv[tid] = s;
    __syncthreads();
    for (int off = 512; off > 0; off >>= 1) {
        if (tid < off) sv[tid] += sv[tid + off];
        __syncthreads();
    }
    const float xnc = fmaxf(sqrtf(sv[0]), EPS);
    __syncthreads();

    // argmax with lowest-index tie-break
    float bv = -INFINITY;
    int   bi = 0;
    for (int i = tid; i < CAP; i += 1024) {
        const float v = sims[i];
        if (v > bv) { bv = v; bi = i; }
    }
    sv[tid] = bv;
    si[tid] = bi;
    __syncthreads();
    for (int off = 512; off > 0; off >>= 1) {
        if (tid < off) {
            const float ov = sv[tid + off];
            const int   oi = si[tid + off];
            if (ov > sv[tid] || (ov == sv[tid] && oi < si[tid])) {
                sv[tid] = ov;
                si[tid] = oi;
            }
        }
        __syncthreads();
    }
    if (tid == 0) {
        ((int*)res)[0] = si[0];
        res[1] = sv[0] / xnc;
    }
}

// ---------------------------------------------------------------------------
// Kernel 3: write the one-hot masked output (rewrites all of d_out; no reads
// of d_out, so it is safe that kernel 1/2 used d_out as the sims scratch).
// ---------------------------------------------------------------------------
__global__ __launch_bounds__(256) void writeout_kernel(
    const float* __restrict__ res,
    float* __restrict__ out)
{
    const int i   = blockIdx.x * blockDim.x + threadIdx.x;
    const int idx = ((const int*)res)[0];
    const float v = res[1];
    out[i] = (i == idx) ? v : 0.f;
}

extern "C" void kernel_launch(void* const* d_in, const int* in_sizes, int n_in,
                              void* d_out, int out_size, void* d_ws, size_t ws_size,
                              hipStream_t stream)
{
    const float* x   = (const float*)d_in[0];   // [4096]
    const float* mem = (const float*)d_in[1];   // [65536, 4096]
    float* out = (float*)d_out;                 // [65536]
    float* res = (float*)d_ws;                  // 2 floats of scratch

    // Pass 1: sims' -> d_out (scratch reuse), one streaming pass over memory.
    gemv_cos_kernel<<<CAP / (16 * WAVES), TPB, 0, stream>>>(x, mem, out);
    // Pass 2: ||x|| + argmax -> ws.
    argmax_kernel<<<1, 1024, 0, stream>>>(x, out, res);
    // Pass 3: one-hot output.
    writeout_kernel<<<CAP / 256, 256, 0, stream>>>(res, out);
}